// HeteroGraphSAGE_60902636257762
// MI455X (gfx1250) — compile-verified
//
#include <hip/hip_runtime.h>
#include <math.h>

// Problem constants (match reference file)
#define NN   20000   // nodes per type
#define EE   320000  // edges per relation
#define DIN  128
#define HH   256
#define GG   32
#define NREL 6
#define NT   4       // 16-wide N-tiles per wave (wave computes 16x64 slab)

typedef __attribute__((ext_vector_type(2))) float v2f;
typedef __attribute__((ext_vector_type(8))) float v8f;

// ---------------------------------------------------------------------------
// WMMA fp32 GEMM: Y[M,N] = X[M,K] @ W[K,N]  (+bias[N] if bias!=0, or +Y if accFlag)
// Each wave computes a 16(M) x 64(N) slab. Per k-step: issue the A b64 load and
// all 8 B b32 loads into distinct registers FIRST, then the 4 WMMAs, so the
// loads drain with staggered s_wait_loadcnt instead of a serial load->wait->wmma
// chain. 4 independent accumulator chains hide V_WMMA_F32_16X16X4_F32 latency.
// grid.x = M/16, grid.y = N/256, block = 128 threads (4 waves -> 256 cols).
// ---------------------------------------------------------------------------
__global__ __launch_bounds__(128)
void wmma_gemm_f32(const float* __restrict__ X, const float* __restrict__ W,
                   const float* __restrict__ bias, float* __restrict__ Y,
                   int M, int K, int N, int accFlag)
{
    const int wave = threadIdx.x >> 5;
    const int lane = threadIdx.x & 31;
    const int half = lane >> 4;   // 0: lanes 0-15, 1: lanes 16-31
    const int idx  = lane & 15;
    const int m0 = blockIdx.x << 4;
    const int n0 = (blockIdx.y * 4 + wave) << 6;   // 64-wide slab per wave
    if (m0 >= M || n0 >= N) return;

    // A fragment: lane half h supplies K = 2h, 2h+1 for row m0+idx
    const float* __restrict__ arow  = X + (size_t)(m0 + idx) * K + 2 * half;
    // B fragment: lane half h supplies rows K = 2h, 2h+1 for col n0+idx(+16t)
    const float* __restrict__ bbase = W + (size_t)(2 * half) * N + (n0 + idx);
    // C/D: VGPR j <-> row m0 + j + 8*half, col n0+idx(+16t)
    float* __restrict__ ybase = Y + (size_t)(m0 + 8 * half) * N + (n0 + idx);

    v8f c[NT];
    if (accFlag) {
#pragma unroll
        for (int t = 0; t < NT; ++t)
#pragma unroll
            for (int j = 0; j < 8; ++j) c[t][j] = ybase[(size_t)j * N + 16 * t];
    } else if (bias) {
#pragma unroll
        for (int t = 0; t < NT; ++t) {
            float bv = bias[n0 + idx + 16 * t];
#pragma unroll
            for (int j = 0; j < 8; ++j) c[t][j] = bv;
        }
    } else {
#pragma unroll
        for (int t = 0; t < NT; ++t)
#pragma unroll
            for (int j = 0; j < 8; ++j) c[t][j] = 0.0f;
    }

#pragma unroll 4
    for (int k = 0; k < K; k += 4) {
        __builtin_prefetch(arow + k + 128, 0, 0);   // global_prefetch_b8, stream A ahead
        v2f a = *(const v2f*)(arow + k);
        v2f b[NT];                                   // all B frags live simultaneously
#pragma unroll
        for (int t = 0; t < NT; ++t) {
            b[t].x = bbase[(size_t)(k + 0) * N + 16 * t];
            b[t].y = bbase[(size_t)(k + 1) * N + 16 * t];
        }
#pragma unroll
        for (int t = 0; t < NT; ++t)
            c[t] = __builtin_amdgcn_wmma_f32_16x16x4_f32(false, a, false, b[t],
                                                         (short)0, c[t], false, false);
    }
#pragma unroll
    for (int t = 0; t < NT; ++t)
#pragma unroll
        for (int j = 0; j < 8; ++j) ybase[(size_t)j * N + 16 * t] = c[t][j];
}

// ---------------------------------------------------------------------------
// Helpers
// ---------------------------------------------------------------------------
__global__ void fill_f32(float* __restrict__ p, long n, float v)
{
    long i = (long)blockIdx.x * blockDim.x + threadIdx.x;
    long stride = (long)gridDim.x * blockDim.x;
    for (; i < n; i += stride) p[i] = v;
}

__global__ void fill_u32(unsigned* __restrict__ p, long n, unsigned v)
{
    long i = (long)blockIdx.x * blockDim.x + threadIdx.x;
    long stride = (long)gridDim.x * blockDim.x;
    for (; i < n; i += stride) p[i] = v;
}

// Pre-sum W_r / b_l over the 3 relations sharing each destination type.
// grid = (HH*HH/256, 2 types, 2 layers), block = 256
__global__ void combine_wr(const float* __restrict__ W_r, const float* __restrict__ b_l,
                           float* __restrict__ wr_sum, float* __restrict__ b_sum)
{
    int i = blockIdx.z;
    int t = blockIdx.y;
    int kn = blockIdx.x * blockDim.x + threadIdx.x;
    int r0, r1, r2;
    if (t == 0) { r0 = 0; r1 = 1; r2 = 3; } else { r0 = 2; r1 = 4; r2 = 5; }
    size_t wb = (size_t)i * NREL * HH * HH;
    wr_sum[((size_t)i * 2 + t) * HH * HH + kn] =
        W_r[wb + (size_t)r0 * HH * HH + kn] +
        W_r[wb + (size_t)r1 * HH * HH + kn] +
        W_r[wb + (size_t)r2 * HH * HH + kn];
    if (kn < HH) {
        size_t bb = (size_t)i * NREL * HH;
        b_sum[((size_t)i * 2 + t) * HH + kn] =
            b_l[bb + r0 * HH + kn] + b_l[bb + r1 * HH + kn] + b_l[bb + r2 * HH + kn];
    }
}

__global__ void degree_kernel(const int* __restrict__ dst, unsigned* __restrict__ deg, int E)
{
    int e = blockIdx.x * blockDim.x + threadIdx.x;
    if (e < E) atomicAdd(&deg[dst[e]], 1u);
}

// block = 256 (one thread per channel), grid-stride over edges
__global__ void scatter_add_rows(const int* __restrict__ src, const int* __restrict__ dst,
                                 const float* __restrict__ Xs, float* __restrict__ Sum, int E)
{
    int c = threadIdx.x;
    for (int e = blockIdx.x; e < E; e += gridDim.x) {
        int s = src[e];
        int d = dst[e];
        atomicAdd(&Sum[(size_t)d * HH + c], Xs[(size_t)s * HH + c]);
    }
}

__global__ void div_deg(float* __restrict__ Sum, const unsigned* __restrict__ deg, int n)
{
    int c = threadIdx.x;
    for (int r = blockIdx.x; r < n; r += gridDim.x) {
        unsigned dcount = deg[r];
        float inv = dcount ? 1.0f / (float)dcount : 0.0f;   // matches / max(cnt,1)
        Sum[(size_t)r * HH + c] *= inv;
    }
}

// Per-channel sum & sum-of-squares: register accumulate, one atomic pair per block.
__global__ void bn_stats(const float* __restrict__ A, float* __restrict__ sums,
                         float* __restrict__ sumsq, int M, float scale)
{
    int c = threadIdx.x;
    float s = 0.f, q = 0.f;
    for (int r = blockIdx.x; r < M; r += gridDim.x) {
        float v = A[(size_t)r * HH + c] * scale;
        s += v; q += v * v;
    }
    atomicAdd(&sums[c], s);
    atomicAdd(&sumsq[c], q);
}

// Out may alias Acc (in-place); Prev==null on layer 0 (no residual)
__global__ void bn_apply(const float* Acc, const float* __restrict__ sums,
                         const float* __restrict__ sumsq,
                         const float* __restrict__ gamma, const float* __restrict__ beta,
                         const float* __restrict__ Prev, float* Out, int M, float scale)
{
    int c = threadIdx.x;
    float mean = sums[c] / (float)M;
    float var  = sumsq[c] / (float)M - mean * mean;   // biased, matches reference
    float inv  = rsqrtf(var + 1e-5f);
    float g = gamma[c], b = beta[c];
    for (int r = blockIdx.x; r < M; r += gridDim.x) {
        float v = (Acc[(size_t)r * HH + c] * scale - mean) * inv * g + b;
        v = fmaxf(v, 0.f);
        if (Prev) v += Prev[(size_t)r * HH + c];
        Out[(size_t)r * HH + c] = v;
    }
}

__global__ void pool_count(const int* __restrict__ batch, unsigned* __restrict__ cnt, int n)
{
    int i = blockIdx.x * blockDim.x + threadIdx.x;
    if (i < n) atomicAdd(&cnt[batch[i]], 1u);
}

__global__ void pool_sum(const float* __restrict__ X, const int* __restrict__ batch,
                         float* __restrict__ gsum, int n)
{
    int c = threadIdx.x;
    for (int r = blockIdx.x; r < n; r += gridDim.x)
        atomicAdd(&gsum[(size_t)batch[r] * HH + c], X[(size_t)r * HH + c]);
}

// grid = GG graphs, block = 256; dot of [2H] concat with W_out, then sigmoid.
__global__ void final_kernel(const float* __restrict__ gsv, const unsigned* __restrict__ cv,
                             const float* __restrict__ gsp, const unsigned* __restrict__ cp,
                             const float* __restrict__ W_out, const float* __restrict__ b_out,
                             float* __restrict__ out)
{
    __shared__ float red[256];
    int g = blockIdx.x, c = threadIdx.x;
    float nv = fmaxf((float)cv[g], 1.f);
    float np = fmaxf((float)cp[g], 1.f);
    float val = (gsv[(size_t)g * HH + c] / nv) * W_out[c] +
                (gsp[(size_t)g * HH + c] / np) * W_out[HH + c];
    red[c] = val;
    __syncthreads();
    for (int s = 128; s > 0; s >>= 1) {
        if (c < s) red[c] += red[c + s];
        __syncthreads();
    }
    if (c == 0) out[g] = 1.0f / (1.0f + __expf(-(red[0] + b_out[0])));
}

// ---------------------------------------------------------------------------
// Host orchestration
// ---------------------------------------------------------------------------
extern "C" void kernel_launch(void* const* d_in, const int* in_sizes, int n_in,
                              void* d_out, int out_size, void* d_ws, size_t ws_size,
                              hipStream_t stream)
{
    const float* x_vuln  = (const float*)d_in[0];
    const float* x_patch = (const float*)d_in[1];
    const float* W_emb   = (const float*)d_in[2];
    const float* b_emb   = (const float*)d_in[3];
    const float* W_l     = (const float*)d_in[4];
    const float* b_l     = (const float*)d_in[5];
    const float* W_r     = (const float*)d_in[6];
    const float* gamma   = (const float*)d_in[7];
    const float* beta    = (const float*)d_in[8];
    const float* W_out   = (const float*)d_in[9];
    const float* b_out   = (const float*)d_in[10];
    const int* edge_index  = (const int*)d_in[11];
    const int* batch_vuln  = (const int*)d_in[12];
    const int* batch_patch = (const int*)d_in[13];
    (void)in_sizes; (void)n_in; (void)out_size; (void)ws_size;

    // workspace carving (256B aligned)
    char* wsp = (char*)d_ws;
    auto carve = [&](size_t bytes) -> char* {
        char* p = wsp;
        wsp += (bytes + 255) & ~(size_t)255;
        return p;
    };
    const size_t act_bytes = (size_t)NN * HH * sizeof(float);
    float* xs0  = (float*)carve(act_bytes);          // xs type 0 (ping)
    float* xs1  = (float*)carve(act_bytes);          // xs type 1 (ping)
    float* acc0 = (float*)carve(act_bytes);          // acc type 0 (pong, BN in-place)
    float* acc1 = (float*)carve(act_bytes);          // acc type 1 (pong)
    float* agg  = (float*)carve(act_bytes);          // per-relation segment-mean buffer
    unsigned* deg = (unsigned*)carve((size_t)NREL * NN * sizeof(unsigned));
    float* wr_sum = (float*)carve((size_t)2 * 2 * HH * HH * sizeof(float));
    float* b_sum  = (float*)carve((size_t)2 * 2 * HH * sizeof(float));
    float* bn_s   = (float*)carve(2 * HH * sizeof(float));   // [sum | sumsq]
    float* psum   = (float*)carve((size_t)2 * GG * HH * sizeof(float));
    unsigned* pcnt = (unsigned*)carve((size_t)2 * GG * sizeof(unsigned));

    const int REL_SRC[NREL] = {0, 0, 0, 1, 1, 1};
    const int REL_DST[NREL] = {0, 0, 1, 0, 1, 1};

    const dim3 gemm_grid(NN / 16, HH / 256);   // each block covers full N=256

    // 0) pre-sum W_r / b_l per dst type; degree counts (relation-only, layer-invariant)
    combine_wr<<<dim3(HH * HH / 256, 2, 2), 256, 0, stream>>>(W_r, b_l, wr_sum, b_sum);
    fill_u32<<<1024, 256, 0, stream>>>(deg, (long)NREL * NN, 0u);
    for (int r = 0; r < NREL; ++r) {
        const int* dstp = edge_index + ((size_t)r * 2 + 1) * EE;
        degree_kernel<<<(EE + 255) / 256, 256, 0, stream>>>(dstp, deg + (size_t)r * NN, EE);
    }

    // 1) input embedding:  xs[t] = x_t @ W_emb[t] + b_emb[t]
    wmma_gemm_f32<<<gemm_grid, 128, 0, stream>>>(x_vuln,  W_emb,                  b_emb,      xs0, NN, DIN, HH, 0);
    wmma_gemm_f32<<<gemm_grid, 128, 0, stream>>>(x_patch, W_emb + (size_t)DIN*HH, b_emb + HH, xs1, NN, DIN, HH, 0);

    // 2) message-passing layers (ping-pong xs <-> acc, BN applied in place on acc)
    float* cur[2] = {xs0, xs1};
    float* nxt[2] = {acc0, acc1};
    for (int i = 0; i < 2; ++i) {
        // self term with combined W_r (seeds acc with summed bias)
        for (int t = 0; t < 2; ++t)
            wmma_gemm_f32<<<gemm_grid, 128, 0, stream>>>(
                cur[t], wr_sum + ((size_t)i * 2 + t) * HH * HH,
                b_sum + ((size_t)i * 2 + t) * HH, nxt[t], NN, HH, HH, 0);
        // neighbor terms: segment-mean gather, then accumulate GEMM
        for (int r = 0; r < NREL; ++r) {
            const int* srcp = edge_index + ((size_t)r * 2 + 0) * EE;
            const int* dstp = edge_index + ((size_t)r * 2 + 1) * EE;
            fill_f32<<<4096, 256, 0, stream>>>(agg, (long)NN * HH, 0.f);
            scatter_add_rows<<<8192, 256, 0, stream>>>(srcp, dstp, cur[REL_SRC[r]], agg, EE);
            div_deg<<<4096, 256, 0, stream>>>(agg, deg + (size_t)r * NN, NN);
            wmma_gemm_f32<<<gemm_grid, 128, 0, stream>>>(
                agg, W_l + ((size_t)i * NREL + r) * HH * HH, nullptr,
                nxt[REL_DST[r]], NN, HH, HH, 1);
        }
        // acc/3 -> BN -> ReLU (+residual for i>0), in place over acc
        for (int t = 0; t < 2; ++t) {
            fill_f32<<<2, 256, 0, stream>>>(bn_s, 2 * HH, 0.f);
            bn_stats<<<128, 256, 0, stream>>>(nxt[t], bn_s, bn_s + HH, NN, 1.0f / 3.0f);
            bn_apply<<<2048, 256, 0, stream>>>(
                nxt[t], bn_s, bn_s + HH,
                gamma + ((size_t)i * 2 + t) * HH, beta + ((size_t)i * 2 + t) * HH,
                (i > 0) ? cur[t] : nullptr, nxt[t], NN, 1.0f / 3.0f);
        }
        // swap ping-pong
        for (int t = 0; t < 2; ++t) { float* tmp = cur[t]; cur[t] = nxt[t]; nxt[t] = tmp; }
    }

    // 3) global mean pool per graph + readout
    fill_f32<<<64, 256, 0, stream>>>(psum, (long)2 * GG * HH, 0.f);
    fill_u32<<<1, 64, 0, stream>>>(pcnt, (long)2 * GG, 0u);
    pool_count<<<(NN + 255) / 256, 256, 0, stream>>>(batch_vuln,  pcnt,      NN);
    pool_count<<<(NN + 255) / 256, 256, 0, stream>>>(batch_patch, pcnt + GG, NN);
    pool_sum<<<4096, 256, 0, stream>>>(cur[0], batch_vuln,  psum,                   NN);
    pool_sum<<<4096, 256, 0, stream>>>(cur[1], batch_patch, psum + (size_t)GG * HH, NN);
    final_kernel<<<GG, 256, 0, stream>>>(psum, pcnt, psum + (size_t)GG * HH, pcnt + GG,
                                         W_out, b_out, (float*)d_out);
}